// Time2Vec_2765958939448
// MI455X (gfx1250) — compile-verified
//
#include <hip/hip_runtime.h>

typedef __attribute__((ext_vector_type(2))) float v2f;
typedef __attribute__((ext_vector_type(8))) float v8f;

// Time2Vec: out[n, d*64+e] = (e==0) ? (x[n,d]*W[d,e]+b[d,e])
//                                   : sin(x[n,d]*W[d,e]+b[d,e])
// N = B*L rows, D = 8, E = 64, out row = 512 f32.
//
// Each wave32 handles a 16-row stripe; 32 WMMA 16x16x4_f32 tiles cover the
// 512 output columns. Rank-2 update per tile:
//   K=0 row: A = x[:,d],  B = W[d, 16 cols]
//   K=1 row: A = 1.0,     B = b[d, 16 cols]
//   C = 0 (inline)  ->  D = x*W + bias  straight from the matrix unit.
// LDS W/b arrays are padded with a zero region so lanes 16-31 (the K=2/K=3
// operand slots, which must be zero) fetch zeros with plain ds_load_b32 and
// immediate offsets -- no per-tile predication.

#define WAVES_PER_BLOCK 8
#define ROWS_PER_WAVE   16
#define ROWS_PER_BLOCK  (WAVES_PER_BLOCK * ROWS_PER_WAVE)   // 128

__global__ __launch_bounds__(256) void time2vec_wmma_kernel(
    const float* __restrict__ x,   // [N, 8]
    const float* __restrict__ W,   // [8, 64]
    const float* __restrict__ b,   // [8, 64]
    float* __restrict__ out,       // [N, 512]
    int nrows)
{
    // [0,512)   : data
    // [512,1024): zeros (zero-operand region for lanes 16-31; max tile offset
    //             is 496 so 512+496 < 1024 always lands in zeros)
    __shared__ float sW[1024];
    __shared__ float sB[1024];

    const int tid = threadIdx.x;
    for (int i = tid; i < 1024; i += 256) {
        sW[i] = (i < 512) ? W[i] : 0.0f;
        sB[i] = (i < 512) ? b[i] : 0.0f;
    }
    __syncthreads();

    const int wave = tid >> 5;      // wave32
    const int lane = tid & 31;
    const int l16  = lane & 15;
    const bool lo  = (lane < 16);

    const long n0 = (long)blockIdx.x * ROWS_PER_BLOCK + (long)wave * ROWS_PER_WAVE;
    if (n0 >= nrows) return;

    // A operand: lanes 0-15 carry K=0 (= x row value) and K=1 (= 1.0);
    // lanes 16-31 carry K=2 / K=3 and must be zero.
    float xr[8];
    if (lo) {
        const float4* px = (const float4*)(x + (n0 + lane) * 8);
        float4 x01 = px[0];
        float4 x23 = px[1];
        xr[0] = x01.x; xr[1] = x01.y; xr[2] = x01.z; xr[3] = x01.w;
        xr[4] = x23.x; xr[5] = x23.y; xr[6] = x23.z; xr[7] = x23.w;
    } else {
#pragma unroll
        for (int d = 0; d < 8; ++d) xr[d] = 0.0f;
    }
    // jnp.nan_to_num: NaN -> 0, +/-inf -> +/-FLT_MAX.
#pragma unroll
    for (int d = 0; d < 8; ++d) {
        float v = xr[d];
        v = (v != v) ? 0.0f : v;
        v = fminf(fmaxf(v, -3.4028234663852886e38f), 3.4028234663852886e38f);
        xr[d] = v;
    }

    const float aone   = lo ? 1.0f : 0.0f;   // K=1 row of A (bias multiplier)
    const int   idx0   = lo ? l16 : 512;     // LDS base: data col vs zero region
    const bool  iszero = (l16 == 0);         // hoisted: e==0 column predicate

    float* __restrict__ obase = out + n0 * 512 + (lo ? 0 : 8) * 512 + l16;

#pragma unroll
    for (int d = 0; d < 8; ++d) {
        v2f a;
        a[0] = xr[d];   // K=0 (lanes 0-15) / K=2 = 0 (lanes 16-31)
        a[1] = aone;    // K=1 (lanes 0-15) / K=3 = 0 (lanes 16-31)
#pragma unroll
        for (int s = 0; s < 4; ++s) {
            const int toff = d * 64 + s * 16;        // uniform tile offset
            v2f bb;
            bb[0] = sW[idx0 + toff];   // K=0 row: W  (zeros for lanes 16-31)
            bb[1] = sB[idx0 + toff];   // K=1 row: b  (zeros for lanes 16-31)

            v8f c = {};                // C = 0 (inline); D = x*W + 1*bias
            c = __builtin_amdgcn_wmma_f32_16x16x4_f32(
                    /*neg_a=*/false, a, /*neg_b=*/false, bb,
                    /*c_mod=*/(short)0, c, /*reuse_a=*/false, /*reuse_b=*/false);

            // Phase 1: 8 independent sin chains into 8 distinct registers
            // (batched v_sin pipelines through the TRANS unit, no v_nop, no
            // per-store register-reuse waits).
            float o[8];
#pragma unroll
            for (int r = 0; r < 8; ++r) {
                float v  = c[r];
                float sv = __sinf(v);                 // v_mul + v_sin_f32
                o[r] = ((s == 0) && iszero) ? v : sv; // keep linear for e==0
            }
            // Phase 2: store clause, all-immediate offsets off one address.
#pragma unroll
            for (int r = 0; r < 8; ++r) {
                __builtin_nontemporal_store(o[r], obase + (toff + r * 512));
            }
        }
    }
}

extern "C" void kernel_launch(void* const* d_in, const int* in_sizes, int n_in,
                              void* d_out, int out_size, void* d_ws, size_t ws_size,
                              hipStream_t stream) {
    const float* x = (const float*)d_in[0];   // [B, L, 8]
    const float* W = (const float*)d_in[1];   // [8, 64]
    const float* b = (const float*)d_in[2];   // [8, 64]
    float* out = (float*)d_out;               // [B, L, 512]

    const int nrows  = in_sizes[0] / 8;       // B*L = 131072
    const int blocks = (nrows + ROWS_PER_BLOCK - 1) / ROWS_PER_BLOCK;
    time2vec_wmma_kernel<<<blocks, 256, 0, stream>>>(x, W, b, out, nrows);
}